// RNNFeatureExtractor_88338887344756
// MI455X (gfx1250) — compile-verified
//
#include <hip/hip_runtime.h>
#include <hip/hip_bf16.h>

// ---------------------------------------------------------------------------
// 5-layer bidirectional tanh-RNN, H=5, B=256, T=4096, fp32 throughout.
//  - Per-layer input projection [B*T,10] x [10,10] done with fp32 WMMA
//    (v_wmma_f32_16x16x4_f32, K padded 10->12, N padded 10->16), branch-free
//    fragment construction (zero-padding via register selects, not exec).
//  - Sequential recurrence: one lane per (batch, dir), h[5] in registers,
//    hardware tanh, global prefetch to hide L2 latency.
// ---------------------------------------------------------------------------

#define BATCH 256
#define TLEN  4096
#define HID   5
#define FEAT  10   // 2*HID, concat(fwd,bwd)

typedef float v2f __attribute__((ext_vector_type(2)));
typedef float v8f __attribute__((ext_vector_type(8)));

#if __has_builtin(__builtin_amdgcn_tanhf)
#define FAST_TANH(x) __builtin_amdgcn_tanhf(x)
#else
#define FAST_TANH(x) tanhf(x)
#endif

// ---------------------------------------------------------------------------
// Input-projection GEMM: pre[r][n] = sum_k hin[r][k]*Wih[dir(n)][n%5][k] + bias
// One wave handles M=16 rows. K=10 -> 3 x K=4 WMMA steps (zero padded).
// A 16x4 fp32 layout: lanes 0-15 hold row=lane, K={0,1}; lanes 16-31 hold the
// same rows, K={2,3}.  B 4x16 mirrored.  C/D: VGPR r = row r|r+8, col = lane&15.
// ---------------------------------------------------------------------------
__global__ void __launch_bounds__(256)
proj_wmma_kernel(const float* __restrict__ hin,   // [B*T][10]
                 const float* __restrict__ Wih,   // [2][5][10] (this layer)
                 const float* __restrict__ bih,   // [2][5]
                 const float* __restrict__ bhh,   // [2][5]
                 float* __restrict__ pre)         // [B*T][10]
{
    const int wave = (blockIdx.x * blockDim.x + threadIdx.x) >> 5;
    const int lane = threadIdx.x & 31;
    const int half = lane >> 4;           // 0: K pair {0,1}, 1: K pair {2,3}
    const int ln   = lane & 15;           // A: row-in-tile; B/C/D: column
    const bool valid = (ln < FEAT);       // columns 10..15 are padding
    const long rowBase = (long)wave * 16;

    // ---- A fragments: three aligned 8B loads, zero-pad K=10,11 via select.
    const float* arow = hin + (rowBase + ln) * FEAT + half * 2;
    const v2f a01 = *(const v2f*)(arow);          // K = half*2 + {0,1}
    const v2f a45 = *(const v2f*)(arow + 4);      // K = half*2 + {4,5}
    const v2f a89 = *(const v2f*)(hin + (rowBase + ln) * FEAT + 8); // K={8,9}
    v2f aC2; aC2.x = half ? 0.0f : a89.x; aC2.y = half ? 0.0f : a89.y;

    // ---- B fragments: clamp invalid columns to row 0, zero via select.
    const int lnc  = valid ? ln : 0;
    const int d    = lnc / HID, hh = lnc % HID;
    const float* wrow = Wih + (d * HID + hh) * FEAT + half * 2; // WT[k][ln]=Wih[d][hh][k]
    const v2f w01 = *(const v2f*)(wrow);
    const v2f w45 = *(const v2f*)(wrow + 4);
    const v2f w89 = *(const v2f*)(Wih + (d * HID + hh) * FEAT + 8);
    v2f b0, b1, b2;
    b0.x = valid ? w01.x : 0.0f;              b0.y = valid ? w01.y : 0.0f;
    b1.x = valid ? w45.x : 0.0f;              b1.y = valid ? w45.y : 0.0f;
    b2.x = (valid && !half) ? w89.x : 0.0f;   b2.y = (valid && !half) ? w89.y : 0.0f;

    // ---- C accumulator initialized with bias, replicated down the column.
    const float biasRaw = bih[d * HID + hh] + bhh[d * HID + hh];
    const float bias = valid ? biasRaw : 0.0f;
    v8f c;
#pragma unroll
    for (int r = 0; r < 8; ++r) c[r] = bias;

    // ---- Three chained K=4 WMMAs (EXEC untouched up to here).
    c = __builtin_amdgcn_wmma_f32_16x16x4_f32(false, a01, false, b0, (short)0, c, false, false);
    c = __builtin_amdgcn_wmma_f32_16x16x4_f32(false, a45, false, b1, (short)0, c, false, false);
    c = __builtin_amdgcn_wmma_f32_16x16x4_f32(false, aC2, false, b2, (short)0, c, false, false);

    // ---- Store D: VGPR r = tile row (r | r+8), column = ln; cols 0..9 only.
    if (valid) {
#pragma unroll
        for (int r = 0; r < 8; ++r) {
            const long m = rowBase + (half ? (r + 8) : r);
            pre[m * FEAT + ln] = c[r];
        }
    }
}

// ---------------------------------------------------------------------------
// Layer-0 recurrence: input dim 1, projection fused (p = x*wih + bias).
// One lane per (batch, dir). dir==1 consumes time reversed, emits at tm.
// ---------------------------------------------------------------------------
__global__ void __launch_bounds__(32)
rnn_layer0_kernel(const float* __restrict__ x,     // [B][T] (B,1,T flat)
                  const float* __restrict__ Wih0,  // [2][5][1]
                  const float* __restrict__ Whh0,  // [2][5][5]
                  const float* __restrict__ bih0,  // [2][5]
                  const float* __restrict__ bhh0,  // [2][5]
                  float* __restrict__ hout)        // [B*T][10]
{
    const int g = blockIdx.x * blockDim.x + threadIdx.x;
    if (g >= BATCH * 2) return;
    const int b = g >> 1, dir = g & 1;

    float wi[HID], bs[HID], w[HID][HID];
#pragma unroll
    for (int i = 0; i < HID; ++i) {
        wi[i] = Wih0[dir * HID + i];
        bs[i] = bih0[dir * HID + i] + bhh0[dir * HID + i];
#pragma unroll
        for (int j = 0; j < HID; ++j) w[i][j] = Whh0[dir * 25 + i * HID + j];
    }

    float h0 = 0.f, h1 = 0.f, h2 = 0.f, h3 = 0.f, h4 = 0.f;
    const long base = (long)b * TLEN;
    const int step = dir ? -1 : 1;
    int tm = dir ? (TLEN - 1) : 0;

    for (int t = 0; t < TLEN; ++t, tm += step) {
        int tp = tm + step * 16;
        tp = (tp < 0) ? 0 : (tp >= TLEN ? TLEN - 1 : tp);
        __builtin_prefetch(x + base + tp, 0, 0);

        const float xv = x[base + tm];
        float a0 = fmaf(xv, wi[0], bs[0]);
        float a1 = fmaf(xv, wi[1], bs[1]);
        float a2 = fmaf(xv, wi[2], bs[2]);
        float a3 = fmaf(xv, wi[3], bs[3]);
        float a4 = fmaf(xv, wi[4], bs[4]);
        a0 = fmaf(w[0][0],h0,fmaf(w[0][1],h1,fmaf(w[0][2],h2,fmaf(w[0][3],h3,fmaf(w[0][4],h4,a0)))));
        a1 = fmaf(w[1][0],h0,fmaf(w[1][1],h1,fmaf(w[1][2],h2,fmaf(w[1][3],h3,fmaf(w[1][4],h4,a1)))));
        a2 = fmaf(w[2][0],h0,fmaf(w[2][1],h1,fmaf(w[2][2],h2,fmaf(w[2][3],h3,fmaf(w[2][4],h4,a2)))));
        a3 = fmaf(w[3][0],h0,fmaf(w[3][1],h1,fmaf(w[3][2],h2,fmaf(w[3][3],h3,fmaf(w[3][4],h4,a3)))));
        a4 = fmaf(w[4][0],h0,fmaf(w[4][1],h1,fmaf(w[4][2],h2,fmaf(w[4][3],h3,fmaf(w[4][4],h4,a4)))));
        h0 = FAST_TANH(a0); h1 = FAST_TANH(a1); h2 = FAST_TANH(a2);
        h3 = FAST_TANH(a3); h4 = FAST_TANH(a4);

        float* o = hout + (base + tm) * FEAT + dir * HID;
        o[0] = h0; o[1] = h1; o[2] = h2; o[3] = h3; o[4] = h4;
    }
}

// ---------------------------------------------------------------------------
// Layers 1..4 recurrence: pre already holds the projection + biases.
// ---------------------------------------------------------------------------
__global__ void __launch_bounds__(32)
rnn_recur_kernel(const float* __restrict__ pre,   // [B*T][10]
                 const float* __restrict__ Whh,   // [2][5][5] (this layer)
                 float* __restrict__ hout)        // [B*T][10]
{
    const int g = blockIdx.x * blockDim.x + threadIdx.x;
    if (g >= BATCH * 2) return;
    const int b = g >> 1, dir = g & 1;

    float w[HID][HID];
#pragma unroll
    for (int i = 0; i < HID; ++i)
#pragma unroll
        for (int j = 0; j < HID; ++j) w[i][j] = Whh[dir * 25 + i * HID + j];

    float h0 = 0.f, h1 = 0.f, h2 = 0.f, h3 = 0.f, h4 = 0.f;
    const long base = (long)b * TLEN;
    const int step = dir ? -1 : 1;
    int tm = dir ? (TLEN - 1) : 0;

    for (int t = 0; t < TLEN; ++t, tm += step) {
        int tp = tm + step * 16;
        tp = (tp < 0) ? 0 : (tp >= TLEN ? TLEN - 1 : tp);
        __builtin_prefetch(pre + (base + tp) * FEAT + dir * HID, 0, 0);

        const float* p = pre + (base + tm) * FEAT + dir * HID;
        float a0 = p[0], a1 = p[1], a2 = p[2], a3 = p[3], a4 = p[4];
        a0 = fmaf(w[0][0],h0,fmaf(w[0][1],h1,fmaf(w[0][2],h2,fmaf(w[0][3],h3,fmaf(w[0][4],h4,a0)))));
        a1 = fmaf(w[1][0],h0,fmaf(w[1][1],h1,fmaf(w[1][2],h2,fmaf(w[1][3],h3,fmaf(w[1][4],h4,a1)))));
        a2 = fmaf(w[2][0],h0,fmaf(w[2][1],h1,fmaf(w[2][2],h2,fmaf(w[2][3],h3,fmaf(w[2][4],h4,a2)))));
        a3 = fmaf(w[3][0],h0,fmaf(w[3][1],h1,fmaf(w[3][2],h2,fmaf(w[3][3],h3,fmaf(w[3][4],h4,a3)))));
        a4 = fmaf(w[4][0],h0,fmaf(w[4][1],h1,fmaf(w[4][2],h2,fmaf(w[4][3],h3,fmaf(w[4][4],h4,a4)))));
        h0 = FAST_TANH(a0); h1 = FAST_TANH(a1); h2 = FAST_TANH(a2);
        h3 = FAST_TANH(a3); h4 = FAST_TANH(a4);

        float* o = hout + (base + tm) * FEAT + dir * HID;
        o[0] = h0; o[1] = h1; o[2] = h2; o[3] = h3; o[4] = h4;
    }
}

// ---------------------------------------------------------------------------
// out[b][0][0:10] = h_last_layer[b][T-1][0:10]
// ---------------------------------------------------------------------------
__global__ void __launch_bounds__(256)
gather_last_kernel(const float* __restrict__ hin, float* __restrict__ out)
{
    const int i = blockIdx.x * blockDim.x + threadIdx.x;
    if (i < BATCH * FEAT) {
        const int b = i / FEAT, c = i % FEAT;
        out[i] = hin[((long)b * TLEN + (TLEN - 1)) * FEAT + c];
    }
}

// ---------------------------------------------------------------------------
extern "C" void kernel_launch(void* const* d_in, const int* in_sizes, int n_in,
                              void* d_out, int out_size, void* d_ws, size_t ws_size,
                              hipStream_t stream) {
    const float* x    = (const float*)d_in[0];  // [256,1,4096]
    const float* Wih0 = (const float*)d_in[1];  // [2,5,1]
    const float* Whh0 = (const float*)d_in[2];  // [2,5,5]
    const float* bih0 = (const float*)d_in[3];  // [2,5]
    const float* bhh0 = (const float*)d_in[4];  // [2,5]
    const float* Wih  = (const float*)d_in[5];  // [4,2,5,10]
    const float* Whh  = (const float*)d_in[6];  // [4,2,5,5]
    const float* bih  = (const float*)d_in[7];  // [4,2,5]
    const float* bhh  = (const float*)d_in[8];  // [4,2,5]
    float* out = (float*)d_out;

    const long nrows = (long)BATCH * TLEN;          // 1,048,576
    float* bufA = (float*)d_ws;                     // [B*T][10]  layer activations
    float* bufP = bufA + nrows * FEAT;              // [B*T][10]  projections

    // Layer 0: fused projection + recurrence, x -> bufA
    rnn_layer0_kernel<<<16, 32, 0, stream>>>(x, Wih0, Whh0, bih0, bhh0, bufA);

    // Layers 1..4: WMMA projection (bufA -> bufP), then recurrence (bufP -> bufA)
    const int projBlocks = (int)((nrows / 16) * 32 / 256);  // 8192 blocks of 8 waves
    for (int l = 1; l < 5; ++l) {
        const int s = l - 1;
        proj_wmma_kernel<<<projBlocks, 256, 0, stream>>>(
            bufA, Wih + s * 2 * HID * FEAT, bih + s * FEAT, bhh + s * FEAT, bufP);
        rnn_recur_kernel<<<16, 32, 0, stream>>>(bufP, Whh + s * 2 * 25, bufA);
    }

    // Extract last timestep
    gather_last_kernel<<<(BATCH * FEAT + 255) / 256, 256, 0, stream>>>(bufA, out);
}